// AbstractFeature_score_74302934220885
// MI455X (gfx1250) — compile-verified
//
#include <hip/hip_runtime.h>
#include <hip/hip_bf16.h>

typedef __attribute__((ext_vector_type(2))) float v2f;
typedef __attribute__((ext_vector_type(4))) float v4f;
typedef __attribute__((ext_vector_type(8))) float v8f;

// ---------------------------------------------------------------------------
// Phase 1: mat-vec via V_WMMA_F32_16X16X4_F32.
// One wave (32 threads) per 16-row tile. First nHrTiles tiles cover hr@W1,
// the rest cover tail@W2.
//
// A layout (32-bit A 16x4, ISA 7.12.2): lane l holds A[l&15, koff + {0,1}]
// in its two VGPRs, koff = (l>>4)*2.
// B layout (4x16, row striped across lanes within a VGPR): lane 0 holds
// (K=0,N=0),(K=1,N=0); lane 16 holds (K=2,N=0),(K=3,N=0). We place w[k..k+3]
// in column N=0 (lanes 0 and 16) and zero elsewhere via a VALU mask multiply
// (NOT a conditional load -> avoids EXEC-predicated loads in the inner loop),
// so D[:,0] accumulates the dot products.
// D layout: VGPR v, lane 0 -> (M=v, N=0); lane 16 -> (M=8+v, N=0).
// ---------------------------------------------------------------------------
__global__ __launch_bounds__(32)
void matvec_wmma_f32(const float* __restrict__ hr,
                     const float* __restrict__ tail,
                     const float* __restrict__ W,   // [2*C], W1 then W2
                     float* __restrict__ hr_part,   // [B]
                     float* __restrict__ tail_part, // [E]
                     int nHrTiles, int C)
{
    const int t    = blockIdx.x;
    const int lane = threadIdx.x;            // 0..31 (wave32)

    const float* M;
    const float* w;
    float*       out;
    int          rbase;
    if (t < nHrTiles) {
        M = hr;   w = W;     out = hr_part;   rbase = t * 16;
    } else {
        M = tail; w = W + C; out = tail_part; rbase = (t - nHrTiles) * 16;
    }

    const int m    = lane & 15;
    const int koff = (lane >> 4) * 2;        // 0 for lanes 0-15, 2 for 16-31
    const float mskf = (m == 0) ? 1.0f : 0.0f;   // lanes 0/16 carry column N=0
    v2f msk; msk.x = mskf; msk.y = mskf;

    const float* __restrict__ rowp = M + (size_t)(rbase + m) * (size_t)C + koff;
    const float* __restrict__ wp   = w + koff;

    v8f acc = {};
    for (int k = 0; k < C; k += 4) {
        v2f a  = *(const v2f*)(rowp + k);    // global_load_b64 (A operand)
        v2f wk = *(const v2f*)(wp + k);      // global_load_b64 (unconditional)
        v2f bm = wk * msk;                   // zero all columns except N=0
        // D = A(16x4) * B(4x16) + C ; emits v_wmma_f32_16x16x4_f32
        acc = __builtin_amdgcn_wmma_f32_16x16x4_f32(
                  false, a, false, bm, (short)0, acc, false, false);
    }

    if (lane == 0) {
#pragma unroll
        for (int v = 0; v < 8; ++v) out[rbase + v] = acc[v];
    } else if (lane == 16) {
#pragma unroll
        for (int v = 0; v < 8; ++v) out[rbase + 8 + v] = acc[v];
    }
}

// ---------------------------------------------------------------------------
// Phase 2: scores[i,j] = hr_part[i] + tail_part[j] + b.
// Store-bandwidth bound (204.8 MB). v4f (b128) NT stores; each thread loads
// one v4f of tail_part and broadcasts it to ROWS_PER_BLOCK rows.
// ---------------------------------------------------------------------------
#define ROWS_PER_BLOCK 4

__global__ __launch_bounds__(256)
void bcast_add(const float* __restrict__ hr_part,
               const float* __restrict__ tail_part,
               const float* __restrict__ bptr,
               float* __restrict__ out,
               int E4 /* E/4 */, int rowStride /* = gridDim.y */)
{
    const int j4 = blockIdx.x * 256 + (int)threadIdx.x;
    if (j4 >= E4) return;

    const float bias = bptr[0];
    const v4f tv = ((const v4f*)tail_part)[j4];

    v4f* __restrict__ out4 = (v4f*)out;
#pragma unroll
    for (int r = 0; r < ROWS_PER_BLOCK; ++r) {
        const int i = (int)blockIdx.y + r * rowStride;
        const float s = hr_part[i] + bias;            // uniform -> scalar load
        v4f o = tv + s;                               // broadcast add
        __builtin_nontemporal_store(o, out4 + (size_t)i * (size_t)E4 + j4);
    }
}

// Zero-fill trailing tuple element(s) (the Python reference returns (scores, 0)).
__global__ void tail_zero(float* __restrict__ out, long long start, long long count)
{
    long long idx = (long long)blockIdx.x * blockDim.x + threadIdx.x;
    if (idx < count) out[start + idx] = 0.0f;
}

extern "C" void kernel_launch(void* const* d_in, const int* in_sizes, int n_in,
                              void* d_out, int out_size, void* d_ws, size_t ws_size,
                              hipStream_t stream)
{
    const float* hr   = (const float*)d_in[0];  // [B, C]
    const float* tail = (const float*)d_in[1];  // [E, C]
    const float* W    = (const float*)d_in[2];  // [1, 2C]
    const float* b    = (const float*)d_in[3];  // [1]

    const int C = in_sizes[2] / 2;              // 512
    const int B = in_sizes[0] / C;              // 1024
    const int E = in_sizes[1] / C;              // 50000

    // Scratch: hr_part [B] at offset 0, tail_part [E] 16B-aligned after it.
    float* hr_part   = (float*)d_ws;
    size_t tailOff   = ((size_t)B * sizeof(float) + 15) & ~(size_t)15;
    float* tail_part = (float*)((char*)d_ws + tailOff);

    // Phase 1: one wave per 16-row tile (B and E are multiples of 16).
    const int nHrTiles   = B / 16;              // 64
    const int nTailTiles = E / 16;              // 3125
    matvec_wmma_f32<<<nHrTiles + nTailTiles, 32, 0, stream>>>(
        hr, tail, W, hr_part, tail_part, nHrTiles, C);

    // Phase 2: broadcast add, v4f stores (E % 4 == 0).
    const int E4 = E / 4;                       // 12500
    dim3 grid((E4 + 255) / 256, B / ROWS_PER_BLOCK);   // 49 x 256
    bcast_add<<<grid, 256, 0, stream>>>(hr_part, tail_part, b,
                                        (float*)d_out, E4, B / ROWS_PER_BLOCK);

    // Trailing "0" element(s) of the tuple output.
    long long scores = (long long)B * (long long)E;
    long long extra  = (long long)out_size - scores;
    if (extra > 0) {
        int thr = 64;
        int blk = (int)((extra + thr - 1) / thr);
        tail_zero<<<blk, thr, 0, stream>>>((float*)d_out, scores, extra);
    }
}